// SSIM_v2_54228257080148
// MI455X (gfx1250) — compile-verified
//
#include <hip/hip_runtime.h>
#include <hip/hip_bf16.h>
#include <cstdint>
#include <math.h>

typedef float v2f __attribute__((ext_vector_type(2)));
typedef float v8f __attribute__((ext_vector_type(8)));

#define NP     16                 // patches per (v,n) -> WMMA M dimension
#define ROWF   363                // 121*3 floats per patch row (K dimension, padded to 364)
#define TILEF  (NP * ROWF)        // 5808 floats per tile
#define WOFF   0                  // gaussian window table: smem[0..127] (121 used, pad zeroed)
#define GTOFF  128
#define PROFF(v) (128 + (1 + (v)) * TILEF)

#if defined(__HIP_DEVICE_COMPILE__) && __has_builtin(__builtin_amdgcn_global_load_async_to_lds_b128)
#define USE_ASYNC_LDS 1
// builtin signature (from hipcc diagnostic): (v4i addrspace(1)*, v4i addrspace(3)*, imm offset, imm cpol)
typedef int v4i_vs __attribute__((vector_size(16)));
typedef __attribute__((address_space(1))) v4i_vs as1_v4i;
typedef __attribute__((address_space(3))) v4i_vs as3_v4i;
#endif

__device__ __forceinline__ void copy16_to_lds(float* smem, int dstIdx, const float* src) {
#ifdef USE_ASYNC_LDS
  // per-lane 16B async copy: global -> LDS, tracked by ASYNCcnt.
  // int round-trips: AS1 flat address == generic address; AS3 offset == low 32 bits of
  // a generic LDS pointer (ISA 10.2).
  __builtin_amdgcn_global_load_async_to_lds_b128(
      (as1_v4i*)(unsigned long long)(uintptr_t)src,
      (as3_v4i*)(unsigned int)(uintptr_t)(&smem[dstIdx]),
      0, 0);
#else
  *(float4*)(&smem[dstIdx]) = *(const float4*)src;
#endif
}

// One step: K-slice [4s, 4s+4) of the five moment chains.
#define SSIM_STEP(S, BPH, LAST)                                                              \
  {                                                                                          \
    const int jb = 4 * (S) + j2;                                                             \
    float px = P[jb], py = P[jb + 1];                                                        \
    float gx = G[jb], gy = G[jb + 1];                                                        \
    float wx = W[jb / 3], wy = W[(jb + 1) / 3];                                              \
    if ((LAST) && hi) { py = 0.0f; gy = 0.0f; }  /* K padding element j==363 */              \
    float wpx = wx * px, wpy = wy * py;                                                      \
    float wgx = wx * gx, wgy = wy * gy;                                                      \
    v2f a1 = {wpx, wpy};                                                                     \
    v2f a2 = {wgx, wgy};                                                                     \
    v2f a3 = {wpx * px, wpy * py};                                                           \
    v2f a4 = {wgx * gx, wgy * gy};                                                           \
    v2f a5 = {wpx * gx, wpy * gy};                                                           \
    accMu1 = __builtin_amdgcn_wmma_f32_16x16x4_f32(false, a1, false, (BPH), (short)0, accMu1, false, false); \
    accMu2 = __builtin_amdgcn_wmma_f32_16x16x4_f32(false, a2, false, (BPH), (short)0, accMu2, false, false); \
    accXX  = __builtin_amdgcn_wmma_f32_16x16x4_f32(false, a3, false, (BPH), (short)0, accXX,  false, false); \
    accYY  = __builtin_amdgcn_wmma_f32_16x16x4_f32(false, a4, false, (BPH), (short)0, accYY,  false, false); \
    accXY  = __builtin_amdgcn_wmma_f32_16x16x4_f32(false, a5, false, (BPH), (short)0, accXY,  false, false); \
  }

__global__ __launch_bounds__(128) void ssim_wmma_f32_kernel(
    const float* __restrict__ pred,   // [n_view, N, 16, 121, 3]
    const float* __restrict__ gt,     // [N, 16, 121, 3]
    float* __restrict__ out,          // [n_view, N, 16]
    int N)
{
  extern __shared__ float smem[];
  const int tid  = threadIdx.x;
  const int n    = blockIdx.x;
  const int nvw  = (int)(blockDim.x >> 5);   // views == waves per block (4)

  // ---- Gaussian 11x11 window (f64, matches numpy then cast to f32) ----
  if (tid == 0) {
    double g[11], S = 0.0;
#pragma unroll
    for (int i = 0; i < 11; ++i) { double x = (double)(i - 5); g[i] = exp(-(x * x) / 4.5); S += g[i]; }
    double inv = 1.0 / (S * S);
    for (int i = 0; i < 11; ++i)
      for (int j = 0; j < 11; ++j)
        smem[WOFF + i * 11 + j] = (float)(g[i] * g[j] * inv);
    for (int k = 121; k < 128; ++k) smem[WOFF + k] = 0.0f;   // pad (used by j==363 weight lookup)
  }

  // ---- Stage gt tile (shared) + one pred tile per view into LDS ----
  {
    const float* gsrc = gt + (size_t)n * TILEF;
    for (int o = tid; o < TILEF / 4; o += blockDim.x)
      copy16_to_lds(smem, GTOFF + 4 * o, gsrc + 4 * o);
    for (int v = 0; v < nvw; ++v) {
      const float* psrc = pred + ((size_t)v * N + n) * TILEF;
      for (int o = tid; o < TILEF / 4; o += blockDim.x)
        copy16_to_lds(smem, PROFF(v) + 4 * o, psrc + 4 * o);
    }
  }
#ifdef USE_ASYNC_LDS
#if __has_builtin(__builtin_amdgcn_s_wait_asynccnt)
  __builtin_amdgcn_s_wait_asynccnt(0);
#else
  asm volatile("s_wait_asynccnt 0" ::: "memory");
#endif
#endif
  __syncthreads();

  // ---- Per-wave WMMA: M = 16 patches, K = 364 (j = k*3+c), N-cols = channel selector ----
  const int lane = tid & 31;
  const int wv   = tid >> 5;          // view handled by this wave
  const bool hi  = lane >= 16;        // high half-wave holds K = 4s+2, 4s+3
  const int row  = lane & 15;         // patch (M) owned by this lane
  const int col  = lane & 15;         // D column owned by this lane
  const int cm   = col % 3;
  const int j2   = hi ? 2 : 0;

  const float* P = &smem[PROFF(wv) + row * ROWF];
  const float* G = &smem[GTOFF     + row * ROWF];
  const float* W = &smem[WOFF];

  // B[j,n] = (j%3 == n%3): 3-periodic in K-steps -> three resident register pairs
  v2f B0, B1, B2;
  {
    v2f b[3];
#pragma unroll
    for (int ph = 0; ph < 3; ++ph) {
      int fx = (ph + j2) % 3;
      int fy = (ph + j2 + 1) % 3;
      b[ph].x = (fx == cm) ? 1.0f : 0.0f;
      b[ph].y = (fy == cm) ? 1.0f : 0.0f;
    }
    B0 = b[0]; B1 = b[1]; B2 = b[2];
  }

  v8f accMu1 = {}, accMu2 = {}, accXX = {}, accYY = {}, accXY = {};

#pragma unroll 1
  for (int t = 0; t < 30; ++t) {      // steps 0..89, phase = s%3
    const int s0 = 3 * t;
    SSIM_STEP(s0 + 0, B0, false)
    SSIM_STEP(s0 + 1, B1, false)
    SSIM_STEP(s0 + 2, B2, false)
  }
  SSIM_STEP(90, B0, true)             // final step covers j = 360..363 (363 masked)

  // ---- SSIM combine + channel mean (columns 0..2 hold channels 0..2) ----
  const float C1v = 1e-4f;            // 0.01^2
  const float C2v = 9e-4f;            // 0.03^2
  float avg[8];
#pragma unroll
  for (int r = 0; r < 8; ++r) {
    float mu1 = accMu1[r], mu2 = accMu2[r];
    float m11 = mu1 * mu1, m22 = mu2 * mu2, m12 = mu1 * mu2;
    float s1  = accXX[r] - m11;
    float s2  = accYY[r] - m22;
    float s12 = accXY[r] - m12;
    float num = (2.0f * m12 + C1v) * (2.0f * s12 + C2v);
    float den = (m11 + m22 + C1v) * (s1 + s2 + C2v);
    float ss  = num / den;
    // channel mean: lane 0 <- lanes 0,1,2 ; lane 16 <- lanes 16,17,18
    float sum3 = ss + __shfl_down(ss, 1, 32) + __shfl_down(ss, 2, 32);
    avg[r] = sum3 * (1.0f / 3.0f);
  }

  if ((lane & 15) == 0) {
    // lane 0: patches 0..7 (D rows M=0..7), lane 16: patches 8..15
    size_t ob = ((size_t)wv * N + n) * 16 + (hi ? 8 : 0);
#pragma unroll
    for (int r = 0; r < 8; ++r) out[ob + r] = avg[r];
  }
}

extern "C" void kernel_launch(void* const* d_in, const int* in_sizes, int n_in,
                              void* d_out, int out_size, void* d_ws, size_t ws_size,
                              hipStream_t stream) {
  (void)n_in; (void)out_size; (void)d_ws; (void)ws_size;
  const float* pred = (const float*)d_in[0];   // [4, N, 16, 121, 3] f32
  const float* gt   = (const float*)d_in[1];   // [N, 16, 121, 3] f32
  float* out = (float*)d_out;                  // [4, N, 16] f32

  const int N      = in_sizes[1] / TILEF;          // 2048
  const int n_view = in_sizes[0] / in_sizes[1];    // 4

  // LDS: 128 (window) + (n_view+1) tiles of 5808 floats + 4 float pad for the K-tail read
  const size_t smem_bytes = (size_t)(128 + (n_view + 1) * TILEF + 4) * sizeof(float);

  dim3 grid((unsigned)N);
  dim3 block((unsigned)(32 * n_view));             // one wave32 per view
  ssim_wmma_f32_kernel<<<grid, block, smem_bytes, stream>>>(pred, gt, out, N);
}